// LightGCNLayer_8117488189797
// MI455X (gfx1250) — compile-verified
//
#include <hip/hip_runtime.h>

#define TILE      256
#define BUF_BYTES 5120          // 2048 rows(i64) + 2048 cols(i64) + 1024 vals(f32)
#define ROWS_OFF  0
#define COLS_OFF  2048
#define VALS_OFF  4096

// ---- CDNA5 async global->LDS copy (ASYNCcnt), non-temporal (edge stream is read once)
__device__ __forceinline__ void async_cp16(unsigned lds_addr, const void* gaddr) {
  asm volatile("global_load_async_to_lds_b128 %0, %1, off th:TH_LOAD_NT"
               :: "v"(lds_addr), "v"(gaddr) : "memory");
}
__device__ __forceinline__ void wait_async0() {
  asm volatile("s_wait_asynccnt 0x0" ::: "memory");
}
// ---- device-scope fp32 atomic add, no-return (STOREcnt; s_endpgm drains)
__device__ __forceinline__ void atomic_add2(float* p, float x, float y) {
  asm volatile("global_atomic_add_f32 %0, %1, off scope:SCOPE_DEV\n\t"
               "global_atomic_add_f32 %0, %2, off offset:4 scope:SCOPE_DEV"
               :: "v"(p), "v"(x), "v"(y) : "memory");
}

extern "C" __global__ void __launch_bounds__(256)
lightgcn_zero_f32(float* __restrict__ out, long long n) {
  long long i = ((long long)blockIdx.x * blockDim.x + threadIdx.x) * 4;
  if (i + 4 <= n) {
    *(float4*)(out + i) = make_float4(0.f, 0.f, 0.f, 0.f);
  } else {
    for (long long j = i; j < n; ++j) out[j] = 0.f;
  }
}

extern "C" __global__ void __launch_bounds__(256)
lightgcn_spmm_scatter(const float* __restrict__ emb,
                      const long long* __restrict__ rows,
                      const long long* __restrict__ cols,
                      const float* __restrict__ vals,
                      float* __restrict__ out,
                      long long nnz, long long ntiles) {
  __shared__ __align__(16) unsigned char smem[2 * BUF_BYTES];
  // Flat LDS pointer: low 32 bits == workgroup-relative LDS byte offset.
  const unsigned lds_base = (unsigned)(unsigned long long)(const void*)smem;

  const int tid  = threadIdx.x;
  const int lane = tid & 31;                                  // wave32
  const int wave = __builtin_amdgcn_readfirstlane(tid >> 5);  // 8 waves / block

  // Stage one 256-edge tile into LDS buffer `buf` with async b128 copies.
  auto stage = [&](long long tile, int buf) {
    const unsigned b   = lds_base + (unsigned)buf * BUF_BYTES;
    const long long e0 = tile * TILE;
    if (tid < 128) {                       // rows: 2 x i64 per 16B chunk
      long long e = e0 + (long long)tid * 2;
      if (e < nnz) async_cp16(b + ROWS_OFF + tid * 16, rows + e);
    } else {                               // cols: 2 x i64 per 16B chunk
      int u = tid - 128;
      long long e = e0 + (long long)u * 2;
      if (e < nnz) async_cp16(b + COLS_OFF + u * 16, cols + e);
    }
    if (tid < 64) {                        // vals: 4 x f32 per 16B chunk
      long long e = e0 + (long long)tid * 4;
      if (e < nnz) async_cp16(b + VALS_OFF + tid * 16, vals + e);
    }
  };

  long long t = blockIdx.x;
  const long long tstride = gridDim.x;
  if (t < ntiles) stage(t, 0);
  int cur = 0;

  for (; t < ntiles; t += tstride) {
    wait_async0();        // this wave's staged data is in LDS
    __syncthreads();      // everyone's data visible

    long long tn = t + tstride;
    if (tn < ntiles) stage(tn, cur ^ 1);   // prefetch next tile into other buffer

    const unsigned char* bp = smem + cur * BUF_BYTES;
    const long long* s_rows = (const long long*)(bp + ROWS_OFF);
    const long long* s_cols = (const long long*)(bp + COLS_OFF);
    const float*     s_vals = (const float*)   (bp + VALS_OFF);

    long long rem = nnz - t * TILE;
    int cnt = rem < TILE ? (int)rem : TILE;
    int e0  = wave * 32;
    int e1  = (e0 + 32 < cnt) ? e0 + 32 : cnt;

    if (e1 - e0 == 32) {
      // Full tile: 4-edge batches -> 4 coalesced b64 gathers in flight,
      // then 8 fire-and-forget f32 atomics per batch per lane-pair.
      #pragma unroll 1
      for (int e = e0; e < e0 + 32; e += 4) {
        long long c0 = s_cols[e+0], c1 = s_cols[e+1], c2 = s_cols[e+2], c3 = s_cols[e+3];
        const float2 g0 = ((const float2*)(emb + (c0 << 6)))[lane];
        const float2 g1 = ((const float2*)(emb + (c1 << 6)))[lane];
        const float2 g2 = ((const float2*)(emb + (c2 << 6)))[lane];
        const float2 g3 = ((const float2*)(emb + (c3 << 6)))[lane];
        long long r0 = s_rows[e+0], r1 = s_rows[e+1], r2 = s_rows[e+2], r3 = s_rows[e+3];
        float v0 = s_vals[e+0], v1 = s_vals[e+1], v2 = s_vals[e+2], v3 = s_vals[e+3];
        atomic_add2(out + (r0 << 6) + lane * 2, v0 * g0.x, v0 * g0.y);
        atomic_add2(out + (r1 << 6) + lane * 2, v1 * g1.x, v1 * g1.y);
        atomic_add2(out + (r2 << 6) + lane * 2, v2 * g2.x, v2 * g2.y);
        atomic_add2(out + (r3 << 6) + lane * 2, v3 * g3.x, v3 * g3.y);
      }
    } else {
      for (int e = e0; e < e1; ++e) {
        long long c = s_cols[e], r = s_rows[e];
        float v = s_vals[e];
        const float2 g = ((const float2*)(emb + (c << 6)))[lane];
        atomic_add2(out + (r << 6) + lane * 2, v * g.x, v * g.y);
      }
    }
    __syncthreads();      // all waves done reading buf before it is re-staged
    cur ^= 1;
  }
}

extern "C" void kernel_launch(void* const* d_in, const int* in_sizes, int n_in,
                              void* d_out, int out_size, void* d_ws, size_t ws_size,
                              hipStream_t stream) {
  const float*     emb  = (const float*)d_in[0];       // [N_NODES, 64] f32
  const long long* rows = (const long long*)d_in[1];   // [NNZ] i64
  const long long* cols = (const long long*)d_in[2];   // [NNZ] i64
  const float*     vals = (const float*)d_in[3];       // [NNZ] f32
  float* out = (float*)d_out;                          // [N_NODES, 64] f32

  const long long nnz   = in_sizes[1];
  const long long n_out = out_size;

  // 1) zero the (poisoned) output
  long long nvec = (n_out + 3) / 4;
  int zb = (int)((nvec + 255) / 256);
  lightgcn_zero_f32<<<zb, 256, 0, stream>>>(out, n_out);

  // 2) gather + atomic scatter with async-LDS double-buffered edge streaming
  long long ntiles = (nnz + TILE - 1) / TILE;
  int blocks = ntiles < 2048 ? (int)ntiles : 2048;
  if (blocks < 1) blocks = 1;
  lightgcn_spmm_scatter<<<blocks, 256, 0, stream>>>(emb, rows, cols, vals, out, nnz, ntiles);
}